// Subseq_YsYp_36507222016274
// MI455X (gfx1250) — compile-verified
//
#include <hip/hip_runtime.h>
#include <hip/hip_bf16.h>
#include <stdint.h>

// Segment attention pooling (two heads), ragged segments, sorted segment ids.
// Memory-bound (AI ~1.8 FLOP/B vs machine balance ~860 FLOP/B) -> single-pass
// HBM traversal via CDNA5 async global->LDS DMA; per-token two-head scores
// computed with V_WMMA_F32_16X16X4_F32 (A = token rows from LDS, B = weights,
// branchless per-lane B base pointer -> 2x ds_load_b64 + wmma inner loop).

#define C_DIM 256
#define TILE  32
#define TS    260   // padded LDS row stride (floats): bank = (4*row + k) % 64

typedef float v2f __attribute__((ext_vector_type(2)));
typedef float v8f __attribute__((ext_vector_type(8)));

__global__ void seg_bounds_kernel(const int* __restrict__ seg,
                                  int* __restrict__ seg_start, int T, int V) {
  int t = blockIdx.x * blockDim.x + threadIdx.x;
  if (t == 0) seg_start[V] = T;
  if (t < T) {
    int s = seg[t];
    if (t == 0 || seg[t - 1] != s) seg_start[s] = t;
  }
}

__global__ __launch_bounds__(C_DIM) void seg_attn_pool_kernel(
    const float* __restrict__ tokens,
    const float* __restrict__ w_ys,
    const float* __restrict__ w_yp,
    const int* __restrict__ seg_start,
    float* __restrict__ out, int V) {
  __shared__ __align__(16) float tile[TILE * TS];     // ~32.5 KB padded token tile
  __shared__ float wsl[C_DIM], wpl[C_DIM], wzero[C_DIM];
  __shared__ float sc_s[TILE], sc_p[TILE];
  __shared__ float e_s[TILE], e_p[TILE];

  const int v    = blockIdx.x;
  const int tid  = threadIdx.x;
  const int lane = tid & 31;
  const int wave = tid >> 5;

  wsl[tid]   = w_ys[tid];
  wpl[tid]   = w_yp[tid];
  wzero[tid] = 0.f;                  // B columns N>=2 read zeros

  const int seg_lo = seg_start[v];
  const int seg_hi = seg_start[v + 1];

  // Loop-invariant WMMA fragment roles.
  const int m    = lane & 15;          // A row (token) / B col (head)
  const int koff = (lane >> 4) << 1;   // K sub-offset per lane half
  const float* wbase = (m == 0) ? wsl : (m == 1) ? wpl : wzero;  // branchless B row

  // Online (flash-style) softmax state, per head.
  float run_m_s = -__builtin_inff(), run_m_p = -__builtin_inff();
  float run_d_s = 0.f, run_d_p = 0.f;
  float acc_s = 0.f, acc_p = 0.f;   // channel c = tid accumulator

  __syncthreads();  // weights visible

  for (int t0 = seg_lo; t0 < seg_hi; t0 += TILE) {
    const int nt = min(TILE, seg_hi - t0);

    // --- Stage tile rows into LDS with CDNA5 async DMA (16B per lane-copy).
    const int nchunk = nt * (C_DIM / 4);
    for (int i = tid; i < nchunk; i += C_DIM) {
      const int r = i >> 6;          // row, 64 x 16B chunks per row
      const int q = i & 63;          // 16B chunk within row
      uint32_t la = (uint32_t)(uintptr_t)(&tile[r * TS + q * 4]);
      uint64_t ga = (uint64_t)(uintptr_t)(tokens + (size_t)(t0 + r) * C_DIM + q * 4);
      asm volatile("global_load_async_to_lds_b128 %0, %1, off"
                   :: "v"(la), "v"(ga) : "memory");
    }
    asm volatile("s_wait_asynccnt 0" ::: "memory");
    __syncthreads();

    // --- Phase 1: two-head scores via WMMA f32 16x16x4.
    // D[16 tok x 16 heads] = A[16 tok x 4 chan] * B[4 chan x 16 heads], K-loop over C.
    if (tid >= nt && tid < TILE) {   // pad tail rows with -inf scores
      sc_s[tid] = -__builtin_inff();
      sc_p[tid] = -__builtin_inff();
    }
    if (wave * 16 < nt) {            // wave-uniform guard: EXEC all-1s at WMMA
      const float* arow = &tile[(wave * 16 + m) * TS];
      v8f d = {};
#pragma unroll 4
      for (int k0 = 0; k0 < C_DIM; k0 += 4) {
        const int kk = k0 + koff;
        const v2f a = *(const v2f*)(arow + kk);    // conflict-free (padded stride)
        const v2f b = *(const v2f*)(wbase + kk);   // broadcast / 2-lane reads
        d = __builtin_amdgcn_wmma_f32_16x16x4_f32(
                false, a, false, b, (short)0, d, false, false);
      }
      // D layout: lanes 0-15 hold M=0..7 in d[0..7], lanes 16-31 hold M=8..15.
      const int tbase = wave * 16 + ((lane >> 4) << 3);
      if (m == 0) {
#pragma unroll
        for (int r = 0; r < 8; ++r) {
          const int t = tbase + r;
          if (t < nt) sc_s[t] = d[r];
        }
      } else if (m == 1) {
#pragma unroll
        for (int r = 0; r < 8; ++r) {
          const int t = tbase + r;
          if (t < nt) sc_p[t] = d[r];
        }
      }
    }
    __syncthreads();

    // --- Phase 2: online-softmax bookkeeping (redundant per thread, LDS broadcast).
    float mt_s = -__builtin_inff(), mt_p = -__builtin_inff();
#pragma unroll
    for (int i = 0; i < TILE; ++i) {
      mt_s = fmaxf(mt_s, sc_s[i]);
      mt_p = fmaxf(mt_p, sc_p[i]);
    }
    const float nm_s = fmaxf(run_m_s, mt_s);       // finite: each tile has >=1 row
    const float nm_p = fmaxf(run_m_p, mt_p);
    const float rscale_s = __expf(run_m_s - nm_s); // exp(-inf - finite) == 0 first tile
    const float rscale_p = __expf(run_m_p - nm_p);
    if (tid < TILE) {
      e_s[tid] = __expf(sc_s[tid] - nm_s);         // -inf rows -> 0
    } else if (tid < 2 * TILE) {
      e_p[tid - TILE] = __expf(sc_p[tid - TILE] - nm_p);
    }
    __syncthreads();

    float dt_s = 0.f, dt_p = 0.f;
#pragma unroll
    for (int i = 0; i < TILE; ++i) { dt_s += e_s[i]; dt_p += e_p[i]; }
    run_d_s = run_d_s * rscale_s + dt_s;
    run_d_p = run_d_p * rscale_p + dt_p;
    acc_s *= rscale_s;
    acc_p *= rscale_p;

    // --- Phase 3: weighted accumulation from the same LDS tile (no 2nd HBM read).
    for (int i = 0; i < nt; ++i) {
      const float x = tile[i * TS + tid];
      acc_s = fmaf(e_s[i], x, acc_s);
      acc_p = fmaf(e_p[i], x, acc_p);
    }
    run_m_s = nm_s;
    run_m_p = nm_p;
    __syncthreads();  // all reads done before next tile's async writes
  }

  out[(size_t)v * C_DIM + tid]       = acc_s / run_d_s;        // ys
  out[(size_t)(V + v) * C_DIM + tid] = acc_p / run_d_p;        // yp
}

extern "C" void kernel_launch(void* const* d_in, const int* in_sizes, int n_in,
                              void* d_out, int out_size, void* d_ws, size_t ws_size,
                              hipStream_t stream) {
  const float* tokens = (const float*)d_in[0];
  const float* w_ys   = (const float*)d_in[1];
  const float* w_yp   = (const float*)d_in[2];
  const int*   seg    = (const int*)d_in[3];
  const int C = in_sizes[1];          // 256
  const int T = in_sizes[3];          // 600000
  const int V = out_size / (2 * C);   // 50000
  if (C != C_DIM) return;             // kernel specialized to C=256

  int* seg_start = (int*)d_ws;        // (V+1) ints, fully rewritten each call

  seg_bounds_kernel<<<(T + 255) / 256, 256, 0, stream>>>(seg, seg_start, T, V);
  seg_attn_pool_kernel<<<V, C_DIM, 0, stream>>>(tokens, w_ys, w_yp, seg_start,
                                                (float*)d_out, V);
}